// EGNNDynamics_48017734369943
// MI455X (gfx1250) — compile-verified
//
#include <hip/hip_runtime.h>
#include <cstdint>
#include <cstddef>

// ---------------------------------------------------------------------------
// EGNN forward for MI455X (gfx1250), fp32 end-to-end using V_WMMA_F32_16X16X4_F32.
// One wave32 computes a 16-row x 64-col output tile (4 x v8f accumulators).
// Weights are repacked once per launch into WMMA-B-fragment order (v2f pairs)
// so every B fragment is one global_load_b64.
// ---------------------------------------------------------------------------

#define N_NODES 50000
#define N_EDGES 800000
#define EDGE_TILES (N_EDGES / 16)   // 50000
#define NODE_TILES (N_NODES / 16)   // 3125
#define LDSS 68                     // padded LDS row stride (floats): conflict-free

typedef __attribute__((ext_vector_type(2))) float v2f;
typedef __attribute__((ext_vector_type(4))) float v4f;
typedef __attribute__((ext_vector_type(8))) float v8f;

__device__ __forceinline__ v8f wmma4(v2f a, v2f b, v8f c) {
  // D = A(16x4,f32) * B(4x16,f32) + C(16x16,f32)
  return __builtin_amdgcn_wmma_f32_16x16x4_f32(
      /*neg_a=*/false, a, /*neg_b=*/false, b,
      /*c_mod=*/(short)0, c, /*reuse_a=*/false, /*reuse_b=*/false);
}

__device__ __forceinline__ float silu_f(float v) {
  return v / (1.0f + __expf(-v));
}

__device__ __forceinline__ v8f splat8(float v) {
  v8f r = {v, v, v, v, v, v, v, v};
  return r;
}

// ---------------------------------------------------------------------------
// Repack row-major weights [2*K2, N] -> fragment pairs: out[p*N+n] = {w[2p][n], w[2p+1][n]}
// ---------------------------------------------------------------------------
__global__ __launch_bounds__(256) void k_pack(const float* __restrict__ w,
                                              v2f* __restrict__ out, int K2,
                                              int Ncols) {
  int i = blockIdx.x * 256 + threadIdx.x;
  if (i >= K2 * Ncols) return;
  int p = i / Ncols, n = i - p * Ncols;
  v2f v;
  v.x = w[(size_t)(2 * p) * Ncols + n];
  v.y = w[(size_t)(2 * p + 1) * Ncols + n];
  out[i] = v;
}

// ---------------------------------------------------------------------------
// radial / normalized coord-diff per edge
// ---------------------------------------------------------------------------
__global__ __launch_bounds__(256) void k_coord2diff(
    const float* __restrict__ x, const int* __restrict__ row,
    const int* __restrict__ col, float* __restrict__ radial,
    float* __restrict__ cd) {
  int e = blockIdx.x * 256 + threadIdx.x;
  if (e >= N_EDGES) return;
  int r = row[e], c = col[e];
  float dx = x[r * 3 + 0] - x[c * 3 + 0];
  float dy = x[r * 3 + 1] - x[c * 3 + 1];
  float dz = x[r * 3 + 2] - x[c * 3 + 2];
  float rad = dx * dx + dy * dy + dz * dz;
  float inv = 1.0f / (sqrtf(rad + 1e-8f) + 1.0f);  // NORM_CONST = 1
  radial[e] = rad;
  cd[e * 3 + 0] = dx * inv;
  cd[e * 3 + 1] = dy * inv;
  cd[e * 3 + 2] = dz * inv;
}

// ---------------------------------------------------------------------------
// h = h_in[ N,16 ] @ emb_w[16,64] + emb_b   (WMMA, K=16)
// ---------------------------------------------------------------------------
__global__ __launch_bounds__(256) void k_embed(
    const float* __restrict__ hin, const v2f* __restrict__ wp,
    const float* __restrict__ bias, float* __restrict__ hout) {
  int wave = threadIdx.x >> 5, lane = threadIdx.x & 31;
  int tile = blockIdx.x * 8 + wave;
  if (tile >= NODE_TILES) return;
  int n0 = tile * 16;
  int ml = lane & 15, kh = lane >> 4;

  v8f acc[4];
#pragma unroll
  for (int nt = 0; nt < 4; ++nt) acc[nt] = splat8(bias[ml + nt * 16]);

#pragma unroll
  for (int kk = 0; kk < 16; kk += 4) {
    int p = (kk >> 1) + kh;           // pair index = kel/2
    v2f a = *(const v2f*)(hin + (size_t)(n0 + ml) * 16 + (kk + kh * 2));
#pragma unroll
    for (int nt = 0; nt < 4; ++nt) {
      v2f b = wp[(size_t)p * 64 + ml + nt * 16];
      acc[nt] = wmma4(a, b, acc[nt]);
    }
  }
#pragma unroll
  for (int r8 = 0; r8 < 8; ++r8) {
    int node = n0 + kh * 8 + r8;
#pragma unroll
    for (int nt = 0; nt < 4; ++nt)
      hout[(size_t)node * 64 + ml + nt * 16] = acc[nt][r8];
  }
}

// ---------------------------------------------------------------------------
// Edge MLP of one GCL: mij = silu(silu([h[row],h[col],rad,d0] @ w1 + b1) @ w2 + b2)
// then agg[row] += mij / 100 via f32 global atomics.
// ---------------------------------------------------------------------------
__global__ __launch_bounds__(256) void k_edge_mlp(
    const float* __restrict__ h, const float* __restrict__ radial,
    const float* __restrict__ dist0, const int* __restrict__ row,
    const int* __restrict__ col, const v2f* __restrict__ wp1,
    const float* __restrict__ w1tail, const float* __restrict__ b1,
    const v2f* __restrict__ wp2, const float* __restrict__ b2,
    float* __restrict__ agg) {
  __shared__ __align__(16) float lds[8][16 * LDSS];
  int wave = threadIdx.x >> 5, lane = threadIdx.x & 31;
  int tile = blockIdx.x * 8 + wave;
  if (tile >= EDGE_TILES) return;
  int e0 = tile * 16;
  int ml = lane & 15, kh = lane >> 4;
  int ri = row[e0 + ml], ci = col[e0 + ml];

  // ---- layer 1: K = 128 via WMMA + K = 2 (radial, dist0) via VALU ----
  v8f acc[4];
#pragma unroll
  for (int nt = 0; nt < 4; ++nt) acc[nt] = splat8(b1[ml + nt * 16]);

  for (int kk = 0; kk < 128; kk += 4) {
    int kel = kk + kh * 2;
    const float* as = (kel < 64) ? (h + (size_t)ri * 64 + kel)
                                 : (h + (size_t)ci * 64 + (kel - 64));
    v2f a = *(const v2f*)as;
    int p = (kk >> 1) + kh;
#pragma unroll
    for (int nt = 0; nt < 4; ++nt) {
      v2f b = wp1[(size_t)p * 64 + ml + nt * 16];
      acc[nt] = wmma4(a, b, acc[nt]);
    }
  }
#pragma unroll
  for (int r8 = 0; r8 < 8; ++r8) {
    int er = e0 + kh * 8 + r8;
    float rad = radial[er], d0v = dist0[er];
#pragma unroll
    for (int nt = 0; nt < 4; ++nt) {
      int n = ml + nt * 16;
      acc[nt][r8] += rad * w1tail[8192 + n] + d0v * w1tail[8256 + n];  // rows 128,129
    }
  }

  // silu -> LDS tile (D-layout -> row-major) for layer-2 A fragments
  float* t = lds[wave];
#pragma unroll
  for (int nt = 0; nt < 4; ++nt)
#pragma unroll
    for (int r8 = 0; r8 < 8; ++r8)
      t[(kh * 8 + r8) * LDSS + ml + nt * 16] = silu_f(acc[nt][r8]);
  asm volatile("" ::: "memory");  // keep DS store->load order (HW DS is in-order per wave)

  // ---- layer 2: K = 64 ----
  v8f a2[4];
#pragma unroll
  for (int nt = 0; nt < 4; ++nt) a2[nt] = splat8(b2[ml + nt * 16]);
  for (int kk = 0; kk < 64; kk += 4) {
    v2f a = *(const v2f*)(t + ml * LDSS + kk + kh * 2);
    int p = (kk >> 1) + kh;
#pragma unroll
    for (int nt = 0; nt < 4; ++nt) {
      v2f b = wp2[(size_t)p * 64 + ml + nt * 16];
      a2[nt] = wmma4(a, b, a2[nt]);
    }
  }

  // silu + scatter-add into agg (already scaled by 1/NORM_FACTOR)
#pragma unroll
  for (int r8 = 0; r8 < 8; ++r8) {
    int dest = row[e0 + kh * 8 + r8];
#pragma unroll
    for (int nt = 0; nt < 4; ++nt) {
      int n = ml + nt * 16;
      atomicAdd(&agg[(size_t)dest * 64 + n], silu_f(a2[nt][r8]) * 0.01f);
    }
  }
}

// ---------------------------------------------------------------------------
// Node MLP of one GCL: h = h + silu([h,agg] @ w1 + b1) @ w2 + b2   (in-place)
// ---------------------------------------------------------------------------
__global__ __launch_bounds__(256) void k_node_mlp(
    float* __restrict__ h, const float* __restrict__ agg,
    const v2f* __restrict__ wp1, const float* __restrict__ b1,
    const v2f* __restrict__ wp2, const float* __restrict__ b2) {
  __shared__ __align__(16) float lds[8][16 * LDSS];
  int wave = threadIdx.x >> 5, lane = threadIdx.x & 31;
  int tile = blockIdx.x * 8 + wave;
  if (tile >= NODE_TILES) return;
  int n0 = tile * 16;
  int ml = lane & 15, kh = lane >> 4;
  int ni = n0 + ml;

  v8f acc[4];
#pragma unroll
  for (int nt = 0; nt < 4; ++nt) acc[nt] = splat8(b1[ml + nt * 16]);

  for (int kk = 0; kk < 128; kk += 4) {
    int kel = kk + kh * 2;
    const float* as = (kel < 64) ? (h + (size_t)ni * 64 + kel)
                                 : (agg + (size_t)ni * 64 + (kel - 64));
    v2f a = *(const v2f*)as;
    int p = (kk >> 1) + kh;
#pragma unroll
    for (int nt = 0; nt < 4; ++nt) {
      v2f b = wp1[(size_t)p * 64 + ml + nt * 16];
      acc[nt] = wmma4(a, b, acc[nt]);
    }
  }

  float* t = lds[wave];
#pragma unroll
  for (int nt = 0; nt < 4; ++nt)
#pragma unroll
    for (int r8 = 0; r8 < 8; ++r8)
      t[(kh * 8 + r8) * LDSS + ml + nt * 16] = silu_f(acc[nt][r8]);
  asm volatile("" ::: "memory");

  v8f a2[4];
#pragma unroll
  for (int nt = 0; nt < 4; ++nt) a2[nt] = splat8(b2[ml + nt * 16]);
  for (int kk = 0; kk < 64; kk += 4) {
    v2f a = *(const v2f*)(t + ml * LDSS + kk + kh * 2);
    int p = (kk >> 1) + kh;
#pragma unroll
    for (int nt = 0; nt < 4; ++nt) {
      v2f b = wp2[(size_t)p * 64 + ml + nt * 16];
      a2[nt] = wmma4(a, b, a2[nt]);
    }
  }

  // residual (no silu on layer-2 output)
#pragma unroll
  for (int r8 = 0; r8 < 8; ++r8) {
    int node = n0 + kh * 8 + r8;
#pragma unroll
    for (int nt = 0; nt < 4; ++nt) {
      size_t idx = (size_t)node * 64 + ml + nt * 16;
      h[idx] = h[idx] + a2[nt][r8];
    }
  }
}

// ---------------------------------------------------------------------------
// Coordinate MLP: phi = silu(silu(inp@w1+b1)@w2+b2)@w3 ; x[row] += cd*phi/100
// ---------------------------------------------------------------------------
__global__ __launch_bounds__(256) void k_coord_mlp(
    const float* __restrict__ h, const float* __restrict__ radial,
    const float* __restrict__ dist0, const float* __restrict__ cd,
    const int* __restrict__ row, const int* __restrict__ col,
    const v2f* __restrict__ wp1, const float* __restrict__ w1tail,
    const float* __restrict__ b1, const v2f* __restrict__ wp2,
    const float* __restrict__ b2, const float* __restrict__ w3,
    float* __restrict__ xout) {
  __shared__ __align__(16) float lds[8][16 * LDSS];
  int wave = threadIdx.x >> 5, lane = threadIdx.x & 31;
  int tile = blockIdx.x * 8 + wave;
  if (tile >= EDGE_TILES) return;
  int e0 = tile * 16;
  int ml = lane & 15, kh = lane >> 4;
  int ri = row[e0 + ml], ci = col[e0 + ml];

  v8f acc[4];
#pragma unroll
  for (int nt = 0; nt < 4; ++nt) acc[nt] = splat8(b1[ml + nt * 16]);

  for (int kk = 0; kk < 128; kk += 4) {
    int kel = kk + kh * 2;
    const float* as = (kel < 64) ? (h + (size_t)ri * 64 + kel)
                                 : (h + (size_t)ci * 64 + (kel - 64));
    v2f a = *(const v2f*)as;
    int p = (kk >> 1) + kh;
#pragma unroll
    for (int nt = 0; nt < 4; ++nt) {
      v2f b = wp1[(size_t)p * 64 + ml + nt * 16];
      acc[nt] = wmma4(a, b, acc[nt]);
    }
  }
#pragma unroll
  for (int r8 = 0; r8 < 8; ++r8) {
    int er = e0 + kh * 8 + r8;
    float rad = radial[er], d0v = dist0[er];
#pragma unroll
    for (int nt = 0; nt < 4; ++nt) {
      int n = ml + nt * 16;
      acc[nt][r8] += rad * w1tail[8192 + n] + d0v * w1tail[8256 + n];
    }
  }

  float* t = lds[wave];
#pragma unroll
  for (int nt = 0; nt < 4; ++nt)
#pragma unroll
    for (int r8 = 0; r8 < 8; ++r8)
      t[(kh * 8 + r8) * LDSS + ml + nt * 16] = silu_f(acc[nt][r8]);
  asm volatile("" ::: "memory");

  v8f a2[4];
#pragma unroll
  for (int nt = 0; nt < 4; ++nt) a2[nt] = splat8(b2[ml + nt * 16]);
  for (int kk = 0; kk < 64; kk += 4) {
    v2f a = *(const v2f*)(t + ml * LDSS + kk + kh * 2);
    int p = (kk >> 1) + kh;
#pragma unroll
    for (int nt = 0; nt < 4; ++nt) {
      v2f b = wp2[(size_t)p * 64 + ml + nt * 16];
      a2[nt] = wmma4(a, b, a2[nt]);
    }
  }

  // phi_m = sum_n silu(s2[m][n]) * w3[n] : stage weighted values, reduce per row
#pragma unroll
  for (int nt = 0; nt < 4; ++nt) {
    int n = ml + nt * 16;
    float w3n = w3[n];
#pragma unroll
    for (int r8 = 0; r8 < 8; ++r8)
      t[(kh * 8 + r8) * LDSS + n] = silu_f(a2[nt][r8]) * w3n;
  }
  asm volatile("" ::: "memory");

  if (kh == 0) {  // lanes 0..15 each reduce one edge row
    const v4f* rp = (const v4f*)(t + ml * LDSS);  // row start is 16B-aligned (68*4*ml)
    v4f s4 = rp[0];
#pragma unroll
    for (int q = 1; q < 16; ++q) s4 += rp[q];
    float phi = (s4.x + s4.y) + (s4.z + s4.w);
    phi *= 0.01f;  // / NORM_FACTOR
    int er = e0 + ml;
    int dest = row[er];
    atomicAdd(&xout[(size_t)dest * 3 + 0], cd[(size_t)er * 3 + 0] * phi);
    atomicAdd(&xout[(size_t)dest * 3 + 1], cd[(size_t)er * 3 + 1] * phi);
    atomicAdd(&xout[(size_t)dest * 3 + 2], cd[(size_t)er * 3 + 2] * phi);
  }
}

// ---------------------------------------------------------------------------
// out = h[ N,64 ] @ out_w[64,16] + out_b   (WMMA, single 16-col tile)
// ---------------------------------------------------------------------------
__global__ __launch_bounds__(256) void k_out(
    const float* __restrict__ h, const v2f* __restrict__ wp,
    const float* __restrict__ bias, float* __restrict__ out) {
  int wave = threadIdx.x >> 5, lane = threadIdx.x & 31;
  int tile = blockIdx.x * 8 + wave;
  if (tile >= NODE_TILES) return;
  int n0 = tile * 16;
  int ml = lane & 15, kh = lane >> 4;

  v8f acc = splat8(bias[ml]);
  for (int kk = 0; kk < 64; kk += 4) {
    v2f a = *(const v2f*)(h + (size_t)(n0 + ml) * 64 + kk + kh * 2);
    int p = (kk >> 1) + kh;
    v2f b = wp[(size_t)p * 16 + ml];
    acc = wmma4(a, b, acc);
  }
#pragma unroll
  for (int r8 = 0; r8 < 8; ++r8)
    out[(size_t)(n0 + kh * 8 + r8) * 16 + ml] = acc[r8];
}

// ---------------------------------------------------------------------------
// host launcher
// ---------------------------------------------------------------------------
extern "C" void kernel_launch(void* const* d_in, const int* in_sizes, int n_in,
                              void* d_out, int out_size, void* d_ws,
                              size_t ws_size, hipStream_t stream) {
  const float* h_in  = (const float*)d_in[0];
  const float* x_in  = (const float*)d_in[1];
  const int*   row   = (const int*)d_in[2];
  const int*   col   = (const int*)d_in[3];
  const float* emb_w = (const float*)d_in[4];
  const float* emb_b = (const float*)d_in[5];
  const float* out_w = (const float*)d_in[6];
  const float* out_b = (const float*)d_in[7];
  const float* e_w1  = (const float*)d_in[8];
  const float* e_b1  = (const float*)d_in[9];
  const float* e_w2  = (const float*)d_in[10];
  const float* e_b2  = (const float*)d_in[11];
  const float* n_w1  = (const float*)d_in[12];
  const float* n_b1  = (const float*)d_in[13];
  const float* n_w2  = (const float*)d_in[14];
  const float* n_b2  = (const float*)d_in[15];
  const float* q_w1  = (const float*)d_in[16];
  const float* q_b1  = (const float*)d_in[17];
  const float* q_w2  = (const float*)d_in[18];
  const float* q_b2  = (const float*)d_in[19];
  const float* q_w3  = (const float*)d_in[20];

  // ---- workspace layout ----
  float* ws = (float*)d_ws;
  float* h_ws   = ws; ws += (size_t)N_NODES * 64;
  float* agg    = ws; ws += (size_t)N_NODES * 64;
  float* radial = ws; ws += (size_t)N_EDGES;
  float* dist0  = ws; ws += (size_t)N_EDGES;
  float* cd     = ws; ws += (size_t)N_EDGES * 3;
  float* x_ws   = ws; ws += (size_t)N_NODES * 3;
  // packed weights (v2f regions; all preceding sizes are even -> 8B aligned)
  v2f* pk = (v2f*)ws;
  v2f* pk_e_w1 = pk; pk += (size_t)4 * 64 * 64;   // rows 0..127 of each layer
  v2f* pk_e_w2 = pk; pk += (size_t)4 * 32 * 64;
  v2f* pk_n_w1 = pk; pk += (size_t)4 * 64 * 64;
  v2f* pk_n_w2 = pk; pk += (size_t)4 * 32 * 64;
  v2f* pk_q_w1 = pk; pk += (size_t)2 * 64 * 64;
  v2f* pk_q_w2 = pk; pk += (size_t)2 * 32 * 64;
  v2f* pk_emb  = pk; pk += (size_t)8 * 64;
  v2f* pk_outw = pk; pk += (size_t)32 * 16;

  const int edge_blocks = EDGE_TILES / 8;        // 6250 (exact)
  const int node_blocks = (NODE_TILES + 7) / 8;  // 391
  const int e_blocks_1d = (N_EDGES + 255) / 256; // 3125 (exact)

  // ---- repack weights into B-fragment order ----
  auto pack = [&](const float* src, v2f* dst, int K2, int Ncols) {
    int tot = K2 * Ncols;
    k_pack<<<(tot + 255) / 256, 256, 0, stream>>>(src, dst, K2, Ncols);
  };
  for (int l = 0; l < 4; ++l) {
    pack(e_w1 + (size_t)l * 130 * 64, pk_e_w1 + (size_t)l * 64 * 64, 64, 64);
    pack(e_w2 + (size_t)l * 64 * 64,  pk_e_w2 + (size_t)l * 32 * 64, 32, 64);
    pack(n_w1 + (size_t)l * 128 * 64, pk_n_w1 + (size_t)l * 64 * 64, 64, 64);
    pack(n_w2 + (size_t)l * 64 * 64,  pk_n_w2 + (size_t)l * 32 * 64, 32, 64);
  }
  for (int b = 0; b < 2; ++b) {
    pack(q_w1 + (size_t)b * 130 * 64, pk_q_w1 + (size_t)b * 64 * 64, 64, 64);
    pack(q_w2 + (size_t)b * 64 * 64,  pk_q_w2 + (size_t)b * 32 * 64, 32, 64);
  }
  pack(emb_w, pk_emb, 8, 64);
  pack(out_w, pk_outw, 32, 16);

  // x working copy (must not mutate inputs)
  hipMemcpyAsync(x_ws, x_in, (size_t)N_NODES * 3 * sizeof(float),
                 hipMemcpyDeviceToDevice, stream);

  // h = embed(h_in)
  k_embed<<<node_blocks, 256, 0, stream>>>(h_in, pk_emb, emb_b, h_ws);

  for (int b = 0; b < 2; ++b) {
    k_coord2diff<<<e_blocks_1d, 256, 0, stream>>>(x_ws, row, col, radial, cd);
    if (b == 0) {
      // dist0 == radial of the initial coordinates
      hipMemcpyAsync(dist0, radial, (size_t)N_EDGES * sizeof(float),
                     hipMemcpyDeviceToDevice, stream);
    }
    for (int i = 0; i < 2; ++i) {
      int l = b * 2 + i;
      hipMemsetAsync(agg, 0, (size_t)N_NODES * 64 * sizeof(float), stream);
      k_edge_mlp<<<edge_blocks, 256, 0, stream>>>(
          h_ws, radial, dist0, row, col,
          pk_e_w1 + (size_t)l * 64 * 64, e_w1 + (size_t)l * 130 * 64,
          e_b1 + (size_t)l * 64, pk_e_w2 + (size_t)l * 32 * 64,
          e_b2 + (size_t)l * 64, agg);
      k_node_mlp<<<node_blocks, 256, 0, stream>>>(
          h_ws, agg, pk_n_w1 + (size_t)l * 64 * 64, n_b1 + (size_t)l * 64,
          pk_n_w2 + (size_t)l * 32 * 64, n_b2 + (size_t)l * 64);
    }
    k_coord_mlp<<<edge_blocks, 256, 0, stream>>>(
        h_ws, radial, dist0, cd, row, col,
        pk_q_w1 + (size_t)b * 64 * 64, q_w1 + (size_t)b * 130 * 64,
        q_b1 + (size_t)b * 64, pk_q_w2 + (size_t)b * 32 * 64,
        q_b2 + (size_t)b * 64, q_w3 + (size_t)b * 64, x_ws);
  }

  // outputs: h_out [N,16] then x_out [N,3], concatenated flat
  k_out<<<node_blocks, 256, 0, stream>>>(h_ws, pk_outw, out_b, (float*)d_out);
  hipMemcpyAsync((float*)d_out + (size_t)N_NODES * 16, x_ws,
                 (size_t)N_NODES * 3 * sizeof(float),
                 hipMemcpyDeviceToDevice, stream);
}